// SlidingWindowAttention_49074296324240
// MI455X (gfx1250) — compile-verified
//
#include <hip/hip_runtime.h>
#include <cstdint>

typedef float v2f __attribute__((ext_vector_type(2)));
typedef float v8f __attribute__((ext_vector_type(8)));

#define D_MODEL   2048
#define NUM_HEADS 16
#define N_KV_HEADS 4
#define HEAD_DIM  128
#define N_REP     4
#define WINDOW    256
#define BSZ       2
#define TSEQ      2048
#define KVD       (N_KV_HEADS * HEAD_DIM)   // 512

// ---------------------------------------------------------------------------
// CDNA5 async global->LDS staging: GLOBAL_LOAD_ASYNC_TO_LDS_B128 (ASYNCcnt).
// Emitted via inline asm (portable across ROCm/upstream clang builtin arity).
// VDST = LDS byte offset (low 32 bits of the flat shared address, per the
// ISA aperture mapping); VADDR = 64-bit global.
// ---------------------------------------------------------------------------
__device__ __forceinline__ void stage16(const float* g, float* l) {
  const uint32_t lds_off = (uint32_t)(uintptr_t)l;
  asm volatile("global_load_async_to_lds_b128 %0, %1, off"
               :
               : "v"(lds_off), "v"(g)
               : "memory");
}

__device__ __forceinline__ void stage_wait() {
  asm volatile("s_wait_asynccnt 0x0" ::: "memory");
}

// ---------------------------------------------------------------------------
// fp32 WMMA GEMM: C[M,N] = A[M,K] @ B[K,N]
// block = 256 threads (8 waves as 2x4), wave computes 2x2 16x16 subtiles
// (32x32 per wave -> 64x128 block tile), K staged in LDS in chunks of 32.
// ---------------------------------------------------------------------------
__global__ __launch_bounds__(256)
void gemm_f32_wmma(const float* __restrict__ A, const float* __restrict__ B,
                   float* __restrict__ C, int M, int N, int K) {
  __shared__ float As[64][36];    // row stride 144B  -> b128-aligned
  __shared__ float Bs[32][132];   // row stride 528B  -> b128-aligned

  const int tid  = threadIdx.x;
  const int lane = tid & 31;
  const int w    = tid >> 5;
  const int wr   = w & 1;         // wave row 0..1  (32 rows each)
  const int wc   = w >> 1;        // wave col 0..3  (32 cols each)
  const int hi   = lane >> 4;
  const int ln   = lane & 15;
  const int m0   = blockIdx.y * 64;
  const int n0   = blockIdx.x * 128;

  v8f acc00 = {}, acc01 = {}, acc10 = {}, acc11 = {};

  const int a_row  = tid >> 3;            // 0..31 (+32 on second pass)
  const int a_col  = (tid & 7) * 4;       // 0..28
  const int b_row  = tid >> 5;            // 0..7  (+8,16,24)
  const int b_col  = (tid & 31) * 4;      // 0..124

  for (int kb = 0; kb < K; kb += 32) {
    // stage A 64x32: 512 float4, 2 per thread
#pragma unroll
    for (int e = 0; e < 2; ++e) {
      const int ar = a_row + e * 32;
      stage16(&A[(size_t)(m0 + ar) * K + kb + a_col], &As[ar][a_col]);
    }
    // stage B 32x128: 1024 float4, 4 per thread
#pragma unroll
    for (int e = 0; e < 4; ++e) {
      const int br = b_row + e * 8;
      stage16(&B[(size_t)(kb + br) * N + n0 + b_col], &Bs[br][b_col]);
    }
    if (kb + 32 < K) {
      __builtin_prefetch(&A[(size_t)(m0 + a_row) * K + kb + 32 + a_col], 0, 3);
      __builtin_prefetch(&B[(size_t)(kb + 32 + b_row) * N + n0 + b_col], 0, 3);
    }
    stage_wait();
    __syncthreads();
#pragma unroll
    for (int kk = 0; kk < 8; ++kk) {
      v2f a0, a1, b0, b1;
      // A 16x4 frags: lanes 0-15 hold K={0,1}, lanes 16-31 hold K={2,3}
      a0.x = As[wr * 32 + ln][kk * 4 + 2 * hi];
      a0.y = As[wr * 32 + ln][kk * 4 + 2 * hi + 1];
      a1.x = As[wr * 32 + 16 + ln][kk * 4 + 2 * hi];
      a1.y = As[wr * 32 + 16 + ln][kk * 4 + 2 * hi + 1];
      // B 4x16 frags: VGPR0 = even K row (lanes 16-31 shifted by 2), VGPR1 = +1
      b0.x = Bs[kk * 4 + 2 * hi][wc * 32 + ln];
      b0.y = Bs[kk * 4 + 2 * hi + 1][wc * 32 + ln];
      b1.x = Bs[kk * 4 + 2 * hi][wc * 32 + 16 + ln];
      b1.y = Bs[kk * 4 + 2 * hi + 1][wc * 32 + 16 + ln];
      acc00 = __builtin_amdgcn_wmma_f32_16x16x4_f32(false, a0, false, b0, (short)0, acc00, false, false);
      acc01 = __builtin_amdgcn_wmma_f32_16x16x4_f32(false, a0, false, b1, (short)0, acc01, false, false);
      acc10 = __builtin_amdgcn_wmma_f32_16x16x4_f32(false, a1, false, b0, (short)0, acc10, false, false);
      acc11 = __builtin_amdgcn_wmma_f32_16x16x4_f32(false, a1, false, b1, (short)0, acc11, false, false);
    }
    __syncthreads();
  }
  // C/D layout: VGPR r -> M = r + 8*hi, N = ln
#pragma unroll
  for (int r = 0; r < 8; ++r) {
    const int mloc = r + 8 * hi;
    const size_t row0 = (size_t)(m0 + wr * 32 + mloc);
    const size_t row1 = row0 + 16;
    C[row0 * N + n0 + wc * 32 + ln]      = acc00[r];
    C[row0 * N + n0 + wc * 32 + 16 + ln] = acc01[r];
    C[row1 * N + n0 + wc * 32 + ln]      = acc10[r];
    C[row1 * N + n0 + wc * 32 + 16 + ln] = acc11[r];
  }
}

// ---------------------------------------------------------------------------
// RoPE in-place on [rows][nheads*128]; pairs (i, i+64) within each head.
// ---------------------------------------------------------------------------
__global__ __launch_bounds__(256)
void rope_inplace(float* __restrict__ D, int rows, int nheads) {
  const int per_row = nheads * 64;
  const int idx = blockIdx.x * blockDim.x + threadIdx.x;
  if (idx >= rows * per_row) return;
  const int row = idx / per_row;
  const int r2  = idx - row * per_row;
  const int hh  = r2 >> 6;
  const int i   = r2 & 63;
  const int t   = row & (TSEQ - 1);
  const float inv = __expf(-(float)(2 * i) * (9.210340371976184f / 128.0f));
  const float ang = (float)t * inv;
  float s, c;
  __sincosf(ang, &s, &c);
  float* base = D + (size_t)row * (nheads * HEAD_DIM) + hh * HEAD_DIM;
  const float x0 = base[i], x1 = base[i + 64];
  base[i]      = x0 * c - x1 * s;
  base[i + 64] = x1 * c + x0 * s;
}

// ---------------------------------------------------------------------------
// Sliding-window flash attention, one wave per (b, head, 16-query tile).
// ---------------------------------------------------------------------------
__global__ __launch_bounds__(32)
void attn_swa(const float* __restrict__ Qg, const float* __restrict__ Kg,
              const float* __restrict__ Vg, float* __restrict__ Yg) {
  __shared__ float Qs[16][132];
  __shared__ float Ks[16][132];
  __shared__ float Vs[16][132];
  __shared__ float Ps[16][20];

  const int lane = threadIdx.x;
  const int hi = lane >> 4, ln = lane & 15;
  int bid = blockIdx.x;
  const int qt = bid & (TSEQ / 16 - 1); bid >>= 7;
  const int h  = bid & (NUM_HEADS - 1); bid >>= 4;
  const int b  = bid;
  const int kvh = h / N_REP;
  const int q0  = qt * 16;
  const size_t rowQ = (size_t)(b * TSEQ + q0);

  for (int i = 0; i < 16; ++i)
    stage16(&Qg[(rowQ + i) * D_MODEL + h * HEAD_DIM + 4 * lane], &Qs[i][4 * lane]);
  stage_wait();
  __syncthreads();

  v8f o[8] = {};
  float mrow[8], lrow[8];
#pragma unroll
  for (int r = 0; r < 8; ++r) { mrow[r] = -1e30f; lrow[r] = 0.0f; }

  const float scale = 0.08838834764831845f;  // 1/sqrt(128)
  int jlo = q0 - (WINDOW - 1); if (jlo < 0) jlo = 0;
  const int kt0 = jlo >> 4;

  for (int kt = kt0; kt <= qt; ++kt) {
    const int k0 = kt * 16;
    for (int i = 0; i < 16; ++i) {
      const size_t rk =
          (size_t)(b * TSEQ + k0 + i) * KVD + kvh * HEAD_DIM + 4 * lane;
      stage16(&Kg[rk], &Ks[i][4 * lane]);
      stage16(&Vg[rk], &Vs[i][4 * lane]);
    }
    stage_wait();
    __syncthreads();

    // S(16x16) = Q . K^T over head dim (32 x K=4 WMMA)
    v8f c = {};
#pragma unroll
    for (int kk = 0; kk < 32; ++kk) {
      v2f a, bb;
      a.x  = Qs[ln][kk * 4 + 2 * hi];  a.y  = Qs[ln][kk * 4 + 2 * hi + 1];
      bb.x = Ks[ln][kk * 4 + 2 * hi];  bb.y = Ks[ln][kk * 4 + 2 * hi + 1];
      c = __builtin_amdgcn_wmma_f32_16x16x4_f32(false, a, false, bb,
                                                (short)0, c, false, false);
    }

    // online softmax: row m = r + 8*hi, col n = ln
#pragma unroll
    for (int r = 0; r < 8; ++r) {
      const int i   = q0 + r + 8 * hi;
      const int j   = k0 + ln;
      const int rel = i - j;
      const float sv = (rel >= 0 && rel < WINDOW) ? c[r] * scale : -1e30f;
      float rmax = sv;
#pragma unroll
      for (int d = 1; d < 16; d <<= 1) rmax = fmaxf(rmax, __shfl_xor(rmax, d, 16));
      const float mnew = fmaxf(mrow[r], rmax);
      const float pe   = (sv < -1e29f) ? 0.0f : __expf(sv - mnew);
      float rsum = pe;
#pragma unroll
      for (int d = 1; d < 16; d <<= 1) rsum += __shfl_xor(rsum, d, 16);
      const float alpha = __expf(mrow[r] - mnew);
      mrow[r] = mnew;
      lrow[r] = lrow[r] * alpha + rsum;
#pragma unroll
      for (int nn = 0; nn < 8; ++nn) o[nn][r] *= alpha;
      Ps[r + 8 * hi][ln] = pe;
    }
    __syncthreads();

    // O += P . V (8 col-subtiles x 4 K-steps)
#pragma unroll
    for (int nn = 0; nn < 8; ++nn) {
#pragma unroll
      for (int kk = 0; kk < 4; ++kk) {
        v2f a, bb;
        a.x  = Ps[ln][kk * 4 + 2 * hi];
        a.y  = Ps[ln][kk * 4 + 2 * hi + 1];
        bb.x = Vs[kk * 4 + 2 * hi][nn * 16 + ln];
        bb.y = Vs[kk * 4 + 2 * hi + 1][nn * 16 + ln];
        o[nn] = __builtin_amdgcn_wmma_f32_16x16x4_f32(false, a, false, bb,
                                                      (short)0, o[nn], false, false);
      }
    }
    __syncthreads();
  }

#pragma unroll
  for (int nn = 0; nn < 8; ++nn) {
#pragma unroll
    for (int r = 0; r < 8; ++r) {
      const int mloc = r + 8 * hi;
      Yg[(rowQ + mloc) * D_MODEL + h * HEAD_DIM + nn * 16 + ln] =
          o[nn][r] / lrow[r];
    }
  }
}

// ---------------------------------------------------------------------------
extern "C" void kernel_launch(void* const* d_in, const int* in_sizes, int n_in,
                              void* d_out, int out_size, void* d_ws, size_t ws_size,
                              hipStream_t stream) {
  const float* x  = (const float*)d_in[0];
  const float* Wq = (const float*)d_in[1];
  const float* Wk = (const float*)d_in[2];
  const float* Wv = (const float*)d_in[3];
  const float* Wo = (const float*)d_in[4];
  float* out = (float*)d_out;

  const int M = BSZ * TSEQ;  // 4096 token rows
  float* Qw = (float*)d_ws;                    // M * 2048
  float* Kw = Qw + (size_t)M * D_MODEL;        // M * 512
  float* Vw = Kw + (size_t)M * KVD;            // M * 512
  float* Yw = Vw + (size_t)M * KVD;            // M * 2048

  dim3 blk(256);
  gemm_f32_wmma<<<dim3(D_MODEL / 128, M / 64), blk, 0, stream>>>(x, Wq, Qw, M, D_MODEL, D_MODEL);
  gemm_f32_wmma<<<dim3(KVD / 128,     M / 64), blk, 0, stream>>>(x, Wk, Kw, M, KVD,     D_MODEL);
  gemm_f32_wmma<<<dim3(KVD / 128,     M / 64), blk, 0, stream>>>(x, Wv, Vw, M, KVD,     D_MODEL);
  rope_inplace<<<(M * NUM_HEADS * 64 + 255) / 256, blk, 0, stream>>>(Qw, M, NUM_HEADS);
  rope_inplace<<<(M * N_KV_HEADS * 64 + 255) / 256, blk, 0, stream>>>(Kw, M, N_KV_HEADS);
  attn_swa<<<BSZ * NUM_HEADS * (TSEQ / 16), dim3(32), 0, stream>>>(Qw, Kw, Vw, Yw);
  gemm_f32_wmma<<<dim3(D_MODEL / 128, M / 64), blk, 0, stream>>>(Yw, Wo, out, M, D_MODEL, D_MODEL);
}